// Net_3642132267062
// MI455X (gfx1250) — compile-verified
//
#include <hip/hip_runtime.h>

typedef float v2f __attribute__((ext_vector_type(2)));
typedef float v8f __attribute__((ext_vector_type(8)));

#define F_IN 128
#define HID  16
#define NC   8

// ---------------- utility ----------------
__global__ void k_zero_i32(int* __restrict__ a, int n) {
  int i = blockIdx.x * blockDim.x + threadIdx.x;
  if (i < n) a[i] = 0;
}

// ---------------- CSR build: count, scan, fill ----------------
__global__ void k_count(const int* __restrict__ dst, int* __restrict__ cnt, int E) {
  int e = blockIdx.x * blockDim.x + threadIdx.x;
  if (e < E) atomicAdd(&cnt[dst[e]], 1);
}

// single-block exclusive scan (Hillis-Steele over 1024-chunks with carry)
__global__ void k_scan_excl(const int* __restrict__ cnt, int* __restrict__ start, int N) {
  __shared__ int buf[1024];
  __shared__ int carry;
  int tid = threadIdx.x;
  if (tid == 0) carry = 0;
  __syncthreads();
  for (int base = 0; base < N; base += 1024) {
    int i = base + tid;
    int v = (i < N) ? cnt[i] : 0;
    buf[tid] = v;
    __syncthreads();
    for (int off = 1; off < 1024; off <<= 1) {
      int t = (tid >= off) ? buf[tid - off] : 0;
      __syncthreads();
      buf[tid] += t;
      __syncthreads();
    }
    if (i < N) start[i] = carry + buf[tid] - v;   // exclusive prefix
    __syncthreads();                               // all reads of carry done
    if (tid == 1023) carry += buf[1023];
    __syncthreads();
  }
}

__global__ void k_fill(const int* __restrict__ src, const int* __restrict__ dst,
                       const int* __restrict__ start, int* __restrict__ cursor,
                       int* __restrict__ csrc, int E) {
  int e = blockIdx.x * blockDim.x + threadIdx.x;
  if (e >= E) return;
  int d = dst[e];
  int p = atomicAdd(&cursor[d], 1);               // int atomics only
  csrc[start[d] + p] = src[e];
}

__global__ void k_dinv(const int* __restrict__ cnt, float* __restrict__ dinv, int N) {
  int i = blockIdx.x * blockDim.x + threadIdx.x;
  if (i < N) dinv[i] = rsqrtf((float)cnt[i] + 1.0f);   // +1 self-loop; deg >= 1
}

// ---------------- layer-1 GEMM: H0 = X @ W1  (fp32 WMMA 16x16x4) ----------------
__global__ __launch_bounds__(128) void k_gemm1(const float* __restrict__ X,
                                               const float* __restrict__ W1,
                                               float* __restrict__ H0,
                                               int N, int nTiles) {
  int wave = blockIdx.x * (blockDim.x >> 5) + (threadIdx.x >> 5);
  if (wave >= nTiles) return;                    // wave-uniform: EXEC stays all-1s
  int lane = threadIdx.x & 31;
  int m = lane & 15;                             // M (A rows) == N (B cols) index
  int p = lane >> 4;                             // half-wave selects K pair
  int row  = wave * 16 + m;
  int rowc = row < N ? row : N - 1;              // clamp loads only
  const float* xrow = X  + (size_t)rowc * F_IN + 2 * p;
  const float* wq   = W1 + (2 * p) * HID + m;
  v8f c = {0.f,0.f,0.f,0.f,0.f,0.f,0.f,0.f};
#pragma unroll
  for (int kk = 0; kk < F_IN / 4; ++kk) {
    v2f a, b;
    a.x = xrow[4 * kk];                          // A[m][4kk+2p], A[m][4kk+2p+1]
    a.y = xrow[4 * kk + 1];
    b.x = wq[(4 * kk) * HID];                    // B[4kk+2p][m], B[4kk+2p+1][m]
    b.y = wq[(4 * kk) * HID + HID];
    c = __builtin_amdgcn_wmma_f32_16x16x4_f32(false, a, false, b, (short)0, c, false, false);
  }
  int base = wave * 16;
#pragma unroll
  for (int r = 0; r < 8; ++r) {
    int rr = base + r + 8 * p;                   // C/D layout: row = r + 8*p
    if (rr < N) H0[(size_t)rr * HID + m] = c[r];
  }
}

// ---------------- layer-2 GEMM: Z = H1 @ W2 (W2 zero-padded 16x8 -> 16x16) ----------------
__global__ __launch_bounds__(128) void k_gemm2(const float* __restrict__ H1,
                                               const float* __restrict__ W2,
                                               float* __restrict__ Z,
                                               int N, int nTiles) {
  int wave = blockIdx.x * (blockDim.x >> 5) + (threadIdx.x >> 5);
  if (wave >= nTiles) return;
  int lane = threadIdx.x & 31;
  int m = lane & 15;
  int p = lane >> 4;
  int row  = wave * 16 + m;
  int rowc = row < N ? row : N - 1;
  const float* hrow = H1 + (size_t)rowc * HID + 2 * p;
  bool nb = (m < NC);
  const float* w2q = W2 + (2 * p) * NC + (nb ? m : 0);
  v8f c = {0.f,0.f,0.f,0.f,0.f,0.f,0.f,0.f};
#pragma unroll
  for (int kk = 0; kk < HID / 4; ++kk) {
    v2f a, b;
    a.x = hrow[4 * kk];
    a.y = hrow[4 * kk + 1];
    b.x = nb ? w2q[(4 * kk) * NC]      : 0.0f;   // padded B columns 8..15 are zero
    b.y = nb ? w2q[(4 * kk) * NC + NC] : 0.0f;
    c = __builtin_amdgcn_wmma_f32_16x16x4_f32(false, a, false, b, (short)0, c, false, false);
  }
  if (nb) {
    int base = wave * 16;
#pragma unroll
    for (int r = 0; r < 8; ++r) {
      int rr = base + r + 8 * p;
      if (rr < N) Z[(size_t)rr * NC + m] = c[r];
    }
  }
}

// ---------------- gather aggregation, layer 1 (fused +b1, ReLU) ----------------
// 4 threads per node, one float4 feature chunk each; no float atomics.
__global__ void k_gather16(const int* __restrict__ start, const int* __restrict__ cnt,
                           const int* __restrict__ csrc, const float* __restrict__ dinv,
                           const float* __restrict__ H, const float* __restrict__ b1,
                           float* __restrict__ H1, int N) {
  int t = blockIdx.x * blockDim.x + threadIdx.x;
  int i = t >> 2, q = t & 3;
  if (i >= N) return;
  float di = dinv[i];
  float4 acc = *(const float4*)(H + (size_t)i * HID + q * 4);
  float s2 = di * di;                            // self-loop: dinv[i]^2
  acc.x *= s2; acc.y *= s2; acc.z *= s2; acc.w *= s2;
  int s0 = start[i], n = cnt[i];
  for (int k = 0; k < n; ++k) {
    int s = csrc[s0 + k];
    float w = dinv[s] * di;
    float4 v = *(const float4*)(H + (size_t)s * HID + q * 4);
    acc.x += v.x * w; acc.y += v.y * w; acc.z += v.z * w; acc.w += v.w * w;
  }
  float4 bb = *(const float4*)(b1 + q * 4);
  acc.x = fmaxf(acc.x + bb.x, 0.f);
  acc.y = fmaxf(acc.y + bb.y, 0.f);
  acc.z = fmaxf(acc.z + bb.z, 0.f);
  acc.w = fmaxf(acc.w + bb.w, 0.f);
  *(float4*)(H1 + (size_t)i * HID + q * 4) = acc;
}

// ---------------- gather aggregation, layer 2 (into d_out) ----------------
__global__ void k_gather8(const int* __restrict__ start, const int* __restrict__ cnt,
                          const int* __restrict__ csrc, const float* __restrict__ dinv,
                          const float* __restrict__ Z, float* __restrict__ O, int N) {
  int t = blockIdx.x * blockDim.x + threadIdx.x;
  int i = t >> 1, h = t & 1;
  if (i >= N) return;
  float di = dinv[i];
  float4 acc = *(const float4*)(Z + (size_t)i * NC + h * 4);
  float s2 = di * di;
  acc.x *= s2; acc.y *= s2; acc.z *= s2; acc.w *= s2;
  int s0 = start[i], n = cnt[i];
  for (int k = 0; k < n; ++k) {
    int s = csrc[s0 + k];
    float w = dinv[s] * di;
    float4 v = *(const float4*)(Z + (size_t)s * NC + h * 4);
    acc.x += v.x * w; acc.y += v.y * w; acc.z += v.z * w; acc.w += v.w * w;
  }
  *(float4*)(O + (size_t)i * NC + h * 4) = acc;
}

// ---------------- bias + log_softmax, in place on d_out ----------------
__global__ void k_bias_logsoftmax(float* __restrict__ O, const float* __restrict__ b2, int N) {
  int i = blockIdx.x * blockDim.x + threadIdx.x;
  if (i >= N) return;
  float4 v0 = *(float4*)(O + (size_t)i * NC);
  float4 v1 = *(float4*)(O + (size_t)i * NC + 4);
  float v[NC] = { v0.x + b2[0], v0.y + b2[1], v0.z + b2[2], v0.w + b2[3],
                  v1.x + b2[4], v1.y + b2[5], v1.z + b2[6], v1.w + b2[7] };
  float mx = v[0];
#pragma unroll
  for (int c = 1; c < NC; ++c) mx = fmaxf(mx, v[c]);
  float s = 0.f;
#pragma unroll
  for (int c = 0; c < NC; ++c) s += expf(v[c] - mx);
  float l = mx + logf(s);
#pragma unroll
  for (int c = 0; c < NC; ++c) v[c] -= l;
  *(float4*)(O + (size_t)i * NC)     = make_float4(v[0], v[1], v[2], v[3]);
  *(float4*)(O + (size_t)i * NC + 4) = make_float4(v[4], v[5], v[6], v[7]);
}

extern "C" void kernel_launch(void* const* d_in, const int* in_sizes, int n_in,
                              void* d_out, int out_size, void* d_ws, size_t ws_size,
                              hipStream_t stream) {
  const float* x  = (const float*)d_in[0];
  const int*   ei = (const int*)  d_in[1];
  const float* W1 = (const float*)d_in[2];
  const float* b1 = (const float*)d_in[3];
  const float* W2 = (const float*)d_in[4];
  const float* b2 = (const float*)d_in[5];
  float* out = (float*)d_out;

  const int N = in_sizes[0] / F_IN;
  const int E = in_sizes[1] / 2;
  const int* src = ei;          // edge_index[0] = row = src
  const int* dst = ei + E;      // edge_index[1] = col = dst

  // workspace (all 4-byte elems, 16-elem aligned blocks):
  //   cnt Na | start Na | cursor Na | csrc Ea | dinv Na | h0 16Na | h1 16Na | z 8Na
  //   ~= (44N + E) * 4 bytes  (~30 MB for N=100k, E=3.2M)
  size_t Na = ((size_t)N + 15) & ~(size_t)15;
  size_t Ea = ((size_t)E + 15) & ~(size_t)15;
  int*   cnt    = (int*)d_ws;
  int*   startA = cnt + Na;
  int*   cursor = startA + Na;
  int*   csrc   = cursor + Na;
  float* dinv   = (float*)(csrc + Ea);
  float* h0     = dinv + Na;
  float* h1     = h0 + 16 * Na;
  float* z      = h1 + 16 * Na;

  const int T = 256;
  int nTiles = (N + 15) / 16;
  int gemmBlocks = (nTiles + 3) / 4;   // 4 waves (128 threads) per block

  // CSR build + normalization
  k_zero_i32<<<(N + T - 1) / T, T, 0, stream>>>(cnt, N);
  k_zero_i32<<<(N + T - 1) / T, T, 0, stream>>>(cursor, N);
  k_count   <<<(E + T - 1) / T, T, 0, stream>>>(dst, cnt, E);
  k_scan_excl<<<1, 1024, 0, stream>>>(cnt, startA, N);
  k_fill    <<<(E + T - 1) / T, T, 0, stream>>>(src, dst, startA, cursor, csrc, E);
  k_dinv    <<<(N + T - 1) / T, T, 0, stream>>>(cnt, dinv, N);

  // layer 1: dense transform (WMMA) + gather aggregation (+b1, ReLU fused)
  k_gemm1   <<<gemmBlocks, 128, 0, stream>>>(x, W1, h0, N, nTiles);
  k_gather16<<<(4 * N + T - 1) / T, T, 0, stream>>>(startA, cnt, csrc, dinv, h0, b1, h1, N);

  // layer 2: dense transform (WMMA) + gather aggregation
  k_gemm2   <<<gemmBlocks, 128, 0, stream>>>(h1, W2, z, N, nTiles);
  k_gather8 <<<(2 * N + T - 1) / T, T, 0, stream>>>(startA, cnt, csrc, dinv, z, out, N);

  // bias + log_softmax
  k_bias_logsoftmax<<<(N + T - 1) / T, T, 0, stream>>>(out, b2, N);
}